// DNF_21071109554827
// MI455X (gfx1250) — compile-verified
//
#include <hip/hip_runtime.h>
#include <stdint.h>

typedef __attribute__((ext_vector_type(2))) float v2f;
typedef __attribute__((ext_vector_type(8))) float v8f;

#define B_      16
#define O_      16
#define NPERM_  240
#define NUM_IN_ 160
#define R_      3
#define C_      32
#define RC_     96
#define BP_     (B_ * NPERM_)   // 3840

// d_out layout (floats): nullary(16) | unary(256) | binary(3840) | ak(46080) | ok(96)
#define OUT_NULLARY 0
#define OUT_UNARY   16
#define OUT_BINARY  272
#define OUT_AK      4112
#define OUT_OK      50192

// workspace layout (floats)
#define WS_FD    0
#define WS_FE    (RC_ * NUM_IN_)          // 15360
#define WS_RULES (2 * RC_ * NUM_IN_)      // 30720 ; rules[b][p][r], BP_*3 floats

// ---------------------------------------------------------------------------
// Kernel 1: softmax(and_kernel/T), sigmoid(or_kernel/T), fused coeffs fd/fe
// ---------------------------------------------------------------------------
__global__ void dnf_prep_kernel(const float* __restrict__ and_k,
                                const float* __restrict__ or_k,
                                const float* __restrict__ temp,
                                float* __restrict__ out,
                                float* __restrict__ ws) {
    int tid = blockIdx.x * blockDim.x + threadIdx.x;
    float invT = 1.0f / temp[0];
    if (tid < RC_ * NUM_IN_) {
        float a0 = and_k[tid * 3 + 0] * invT;
        float a1 = and_k[tid * 3 + 1] * invT;
        float a2 = and_k[tid * 3 + 2] * invT;
        float m  = fmaxf(a0, fmaxf(a1, a2));
        float e0 = __expf(a0 - m), e1 = __expf(a1 - m), e2 = __expf(a2 - m);
        float inv = 1.0f / (e0 + e1 + e2);
        e0 *= inv; e1 *= inv; e2 *= inv;
        out[OUT_AK + tid * 3 + 0] = e0;
        out[OUT_AK + tid * 3 + 1] = e1;
        out[OUT_AK + tid * 3 + 2] = e2;
        ws[WS_FD + tid] = e0 - e1;   // x*e0 + (1-x)*e1 + e2 = x*(e0-e1) + (e1+e2)
        ws[WS_FE + tid] = e1 + e2;
    }
    if (tid < RC_) {
        float v = or_k[tid] * invT;
        out[OUT_OK + tid] = 1.0f / (1.0f + __expf(-v));
    }
}

// ---------------------------------------------------------------------------
// Kernel 2: main DNF evaluation. One wave32 per (batch, permutation) pair.
//  - x gather: 5 coalesced 128B rows moved with GLOBAL_LOAD_ASYNC_TO_LDS_B32
//    (ASYNCcnt-tracked, no VGPR round trip).
//  - product over NUM_IN in log domain; K-reduction via V_WMMA_F32_16X16X4_F32
//    with an all-ones B matrix (all D columns hold the log-sum).
// ---------------------------------------------------------------------------
__global__ __launch_bounds__(128)
void dnf_main_kernel(const float* __restrict__ nullary,
                     const float* __restrict__ unary,
                     const float* __restrict__ binary,
                     const float* __restrict__ ws_fd,
                     const float* __restrict__ ws_fe,
                     const float* __restrict__ ok,
                     float* __restrict__ rules) {
    __shared__ float xs[4][NUM_IN_];
    __shared__ float conj[4][RC_];

    const int lane = threadIdx.x & 31;
    const int wave = threadIdx.x >> 5;
    const int bp   = blockIdx.x * 4 + wave;       // 0 .. 3839
    const int b    = bp / NPERM_;
    const int p    = bp % NPERM_;
    const int i    = p / 15;
    const int jj   = p % 15;
    const int j    = jj + (jj >= i);              // itertools.permutations order
    const int colj = j - (j > i);
    const int coli = i - (i > j);

    // --- Async gather of the 160-float input vector into LDS ----------------
    // Segment s covers k in [32s, 32s+32); each lane copies one float.
    const float* srcs[5];
    srcs[0] = nullary + b * 32;
    srcs[1] = unary   + (b * O_ + i) * 32;
    srcs[2] = unary   + (b * O_ + j) * 32;
    srcs[3] = binary  + ((b * O_ + i) * 15 + colj) * 32;
    srcs[4] = binary  + ((b * O_ + j) * 15 + coli) * 32;
#pragma unroll
    for (int s = 0; s < 5; ++s) {
        uint64_t gaddr = (uint64_t)(uintptr_t)(srcs[s] + lane);
        // Flat LDS pointer: addr[31:0] is the LDS byte address (aperture map).
        uint32_t laddr = (uint32_t)(uintptr_t)&xs[wave][s * 32 + lane];
        asm volatile("global_load_async_to_lds_b32 %0, %1, off"
                     :: "v"(laddr), "v"(gaddr) : "memory");
    }
    asm volatile("s_wait_asynccnt 0" ::: "memory");
    __syncthreads();

    const int m    = lane & 15;   // rc-row within tile held by this lane
    const int half = lane >> 4;   // lanes 0-15: K slots 0,1 ; lanes 16-31: 2,3

    v2f onesv; onesv.x = 1.0f; onesv.y = 1.0f;   // B = all-ones 4x16

    for (int tile = 0; tile < 6; ++tile) {
        const int rc = tile * 16 + m;
        const float* __restrict__ fdr = ws_fd + rc * NUM_IN_;
        const float* __restrict__ fer = ws_fe + rc * NUM_IN_;
        if (tile < 5) {  // prefetch next tile's coefficient rows
            __builtin_prefetch(ws_fd + (rc + 16) * NUM_IN_, 0, 0);
            __builtin_prefetch(ws_fe + (rc + 16) * NUM_IN_, 0, 0);
        }

        v8f acc = {};   // log-sum accumulator (C/D of WMMA)
#pragma unroll
        for (int chunk = 0; chunk < 5; ++chunk) {
            // This lane's two K-slots each cover 8 consecutive real k's.
            const int base0 = chunk * 32 + half * 16;
            const int base1 = base0 + 8;
            float p0 = 1.0f, p1 = 1.0f;
#pragma unroll
            for (int g = 0; g < 8; ++g) {
                p0 *= fmaf(xs[wave][base0 + g], fdr[base0 + g], fer[base0 + g]);
                p1 *= fmaf(xs[wave][base1 + g], fdr[base1 + g], fer[base1 + g]);
            }
            v2f a; a.x = __logf(p0); a.y = __logf(p1);
            // D[m, n] += sum_kk A[m, kk]  (ones-B): 16x16x4 f32 WMMA
            acc = __builtin_amdgcn_wmma_f32_16x16x4_f32(
                false, a, false, onesv, (short)0, acc, false, false);
        }
        // Column N=0 of D: lane 0 holds M=0..7, lane 16 holds M=8..15.
        if (m == 0) {
#pragma unroll
            for (int t = 0; t < 8; ++t)
                conj[wave][tile * 16 + half * 8 + t] = __expf(acc[t]);
        }
    }
    __syncthreads();

    // Disjuncts: 1 - prod_c (1 - conj*ok); cross-lane product over C=32 (wave32).
    float dis0 = 0.0f, dis1 = 0.0f, dis2 = 0.0f;
#pragma unroll
    for (int r = 0; r < R_; ++r) {
        float t = 1.0f - conj[wave][r * 32 + lane] * ok[r * 32 + lane];
#pragma unroll
        for (int off = 1; off < 32; off <<= 1)
            t *= __shfl_xor(t, off, 32);
        float d = 1.0f - t;
        if (r == 0) dis0 = d; else if (r == 1) dis1 = d; else dis2 = d;
    }
    if (lane == 0) {
        rules[bp * R_ + 0] = dis0;
        rules[bp * R_ + 1] = dis1;
        rules[bp * R_ + 2] = dis2;
    }
}

// ---------------------------------------------------------------------------
// Kernel 3: probsum reductions over permutations -> final rule outputs
// ---------------------------------------------------------------------------
__global__ void dnf_final_kernel(const float* __restrict__ rules,
                                 float* __restrict__ out) {
    __shared__ float s0[NPERM_];
    __shared__ float s1[NPERM_];
    const int b = blockIdx.x;
    const int p = threadIdx.x;
    if (p < NPERM_) {
        float r0 = rules[(b * NPERM_ + p) * R_ + 0];
        float r1 = rules[(b * NPERM_ + p) * R_ + 1];
        float r2 = rules[(b * NPERM_ + p) * R_ + 2];
        out[OUT_BINARY + b * NPERM_ + p] = r2;   // binary_rules passthrough
        s0[p] = 1.0f - r0;
        s1[p] = 1.0f - r1;
    }
    __syncthreads();
    if (p < O_) {                                 // unary: probsum over 15 perms
        float prod = 1.0f;
        for (int q = 0; q < 15; ++q) prod *= s1[p * 15 + q];
        out[OUT_UNARY + b * O_ + p] = 1.0f - prod;
    }
    if (p == 0) {                                 // nullary: probsum over all 240
        float prod = 1.0f;
        for (int q = 0; q < NPERM_; ++q) prod *= s0[q];
        out[OUT_NULLARY + b] = 1.0f - prod;
    }
}

// ---------------------------------------------------------------------------
extern "C" void kernel_launch(void* const* d_in, const int* in_sizes, int n_in,
                              void* d_out, int out_size, void* d_ws, size_t ws_size,
                              hipStream_t stream) {
    const float* nullary = (const float*)d_in[0];
    const float* unary   = (const float*)d_in[1];
    const float* binary  = (const float*)d_in[2];
    const float* and_k   = (const float*)d_in[3];
    const float* or_k    = (const float*)d_in[4];
    const float* temp    = (const float*)d_in[5];
    float* out = (float*)d_out;
    float* ws  = (float*)d_ws;

    dnf_prep_kernel<<<(RC_ * NUM_IN_ + 255) / 256, 256, 0, stream>>>(
        and_k, or_k, temp, out, ws);

    dnf_main_kernel<<<BP_ / 4, 128, 0, stream>>>(
        nullary, unary, binary,
        ws + WS_FD, ws + WS_FE, out + OUT_OK, ws + WS_RULES);

    dnf_final_kernel<<<B_, 256, 0, stream>>>(ws + WS_RULES, out);
}